// QM9GNN_2104533975162
// MI455X (gfx1250) — compile-verified
//
#include <hip/hip_runtime.h>
#include <hip/hip_bf16.h>

#define NN    100000
#define EE    1600000
#define DIM0  11
#define HH    128
#define LL    4
#define GG    4096
#define EPSV  1e-5f

typedef float v2f __attribute__((ext_vector_type(2)));
typedef float v8f __attribute__((ext_vector_type(8)));

static inline unsigned ceil_div(long a, long b) { return (unsigned)((a + b - 1) / b); }

// ---------------- utility fills ----------------
__global__ void fill_kernel(float* __restrict__ p, float v, long n) {
    long i = (long)blockIdx.x * blockDim.x + threadIdx.x;
    if (i < n) p[i] = v;
}

__global__ void biasinit_kernel(float* __restrict__ acc, const float* __restrict__ bias, long n) {
    long i = (long)blockIdx.x * blockDim.x + threadIdx.x;
    if (i < n) acc[i] = bias[i & (HH - 1)];
}

// ---------------- degree / norm ----------------
__global__ void deg_kernel(const int* __restrict__ dst, float* __restrict__ dis, int nE) {
    int e = blockIdx.x * blockDim.x + threadIdx.x;
    if (e < nE) atomicAdd(&dis[dst[e]], 1.0f);
}

__global__ void rsqrt_kernel(float* __restrict__ dis, int n) {
    int i = blockIdx.x * blockDim.x + threadIdx.x;
    if (i < n) dis[i] = rsqrtf(dis[i]);   // deg >= 1 due to self loop
}

__global__ void norm_kernel(const int* __restrict__ src, const int* __restrict__ dst,
                            const float* __restrict__ dis, float* __restrict__ nrm, int nE) {
    int e = blockIdx.x * blockDim.x + threadIdx.x;
    if (e < nE) nrm[e] = dis[src[e]] * dis[dst[e]];
}

// ---------------- embedding: h = relu(x @ W_embed + b) ----------------
__global__ void embed_kernel(const float* __restrict__ x, const float* __restrict__ We,
                             const float* __restrict__ be, float* __restrict__ h, long total) {
    long i = (long)blockIdx.x * blockDim.x + threadIdx.x;
    if (i >= total) return;
    int n = (int)(i >> 7), c = (int)(i & (HH - 1));
    const float* xr = x + (long)n * DIM0;
    float s = be[c];
#pragma unroll
    for (int d = 0; d < DIM0; ++d) s += xr[d] * We[d * HH + c];
    h[i] = fmaxf(s, 0.0f);
}

// ---------------- dense GEMM: m = h @ W  (f32 WMMA 16x16x4) ----------------
// block: 256 threads = 8 waves; blockIdx.x -> 8 row-tiles (128 rows), blockIdx.y -> col-tile (16 cols)
__global__ __launch_bounds__(256)
void gemm_kernel(const float* __restrict__ h, const float* __restrict__ W,
                 float* __restrict__ m, int nrows) {
    __shared__ float Wl[16 * HH];            // W[:, tileN*16 .. +16), stored K-contiguous per column
    const int tid   = threadIdx.x;
    const int tileN = blockIdx.y;
    for (int j = tid; j < 16 * HH; j += 256) {
        int c = j >> 7, k = j & (HH - 1);
        Wl[j] = W[k * HH + tileN * 16 + c];  // Wl[c*128 + k] = W[k][col]
    }
    __syncthreads();

    const int wave  = tid >> 5;
    const int lane  = tid & 31;
    const int hi    = lane >> 4;             // 0: K pair {0,1}; 1: K pair {2,3}
    const int lm    = lane & 15;
    const int tileM = blockIdx.x * 8 + wave;

    int row  = tileM * 16 + lm;
    int rowc = row < nrows ? row : nrows - 1;        // clamp (no EXEC divergence at WMMA)
    const float* A = h + (size_t)rowc * HH + hi * 2; // per-lane A pointer
    const float* B = Wl + lm * HH + hi * 2;          // per-lane B pointer (LDS)

    v8f acc = {};
#pragma unroll
    for (int k0 = 0; k0 < HH; k0 += 4) {
        v2f a = *(const v2f*)(A + k0);               // global_load_b64
        v2f b = *(const v2f*)(B + k0);               // ds_load_b64
        acc = __builtin_amdgcn_wmma_f32_16x16x4_f32(
                  false, a, false, b, (short)0, acc, false, false);
    }

    const int col = tileN * 16 + lm;
#pragma unroll
    for (int v = 0; v < 8; ++v) {
        int r = tileM * 16 + v + hi * 8;             // D: lanes0-15 M=v, lanes16-31 M=v+8
        if (r < nrows) m[(size_t)r * HH + col] = acc[v];
    }
}

// ---------------- edge scatter: acc[dst] += norm * m[src] ----------------
// one wave per (edge or self-loop); lane handles 4 channels
__global__ __launch_bounds__(256)
void scatter_kernel(const float* __restrict__ m, const int* __restrict__ src,
                    const int* __restrict__ dst, const float* __restrict__ nrm,
                    const float* __restrict__ dis, float* __restrict__ acc,
                    int nE, int nN) {
    long gw = ((long)blockIdx.x * blockDim.x + threadIdx.x) >> 5;
    if (gw >= (long)nE + nN) return;
    int lane = threadIdx.x & 31;
    int s, d; float w;
    if (gw < nE) { s = src[gw]; d = dst[gw]; w = nrm[gw]; }
    else { int n = (int)(gw - nE); s = n; d = n; float r = dis[n]; w = r * r; }
    const float4 v = *(const float4*)(m + (size_t)s * HH + lane * 4);
    float* ap = acc + (size_t)d * HH + lane * 4;
    atomicAdd(ap + 0, v.x * w);
    atomicAdd(ap + 1, v.y * w);
    atomicAdd(ap + 2, v.z * w);
    atomicAdd(ap + 3, v.w * w);
}

// ---------------- layernorm + relu + residual (in-place into h) ----------------
__global__ __launch_bounds__(256)
void ln_kernel(float* __restrict__ h, const float* __restrict__ acc,
               const float* __restrict__ g, const float* __restrict__ b, int nN) {
    long gw = ((long)blockIdx.x * blockDim.x + threadIdx.x) >> 5;
    if (gw >= nN) return;
    int lane = threadIdx.x & 31;
    size_t off = (size_t)gw * HH + lane * 4;
    float4 v = *(const float4*)(acc + off);
    float s = v.x + v.y + v.z + v.w;
#pragma unroll
    for (int o = 16; o >= 1; o >>= 1) s += __shfl_xor(s, o, 32);
    float mean = s * (1.0f / HH);
    float d0 = v.x - mean, d1 = v.y - mean, d2 = v.z - mean, d3 = v.w - mean;
    float q = d0 * d0 + d1 * d1 + d2 * d2 + d3 * d3;
#pragma unroll
    for (int o = 16; o >= 1; o >>= 1) q += __shfl_xor(q, o, 32);
    float inv = rsqrtf(q * (1.0f / HH) + EPSV);
    float4 gv = *(const float4*)(g + lane * 4);
    float4 bv = *(const float4*)(b + lane * 4);
    float4 r  = *(const float4*)(h + off);
    float4 o4;
    o4.x = fmaxf(d0 * inv * gv.x + bv.x, 0.0f) + r.x;
    o4.y = fmaxf(d1 * inv * gv.y + bv.y, 0.0f) + r.y;
    o4.z = fmaxf(d2 * inv * gv.z + bv.z, 0.0f) + r.z;
    o4.w = fmaxf(d3 * inv * gv.w + bv.w, 0.0f) + r.w;
    *(float4*)(h + off) = o4;
}

// ---------------- mean pool (accumulate) ----------------
__global__ __launch_bounds__(256)
void pool_kernel(const float* __restrict__ h, const int* __restrict__ batch,
                 float* __restrict__ gsum, float* __restrict__ gcnt, int nN) {
    long gw = ((long)blockIdx.x * blockDim.x + threadIdx.x) >> 5;
    if (gw >= nN) return;
    int lane = threadIdx.x & 31;
    int g = batch[gw];
    if (lane == 0) atomicAdd(&gcnt[g], 1.0f);
    float4 v = *(const float4*)(h + (size_t)gw * HH + lane * 4);
    float* p = gsum + (size_t)g * HH + lane * 4;
    atomicAdd(p + 0, v.x);
    atomicAdd(p + 1, v.y);
    atomicAdd(p + 2, v.z);
    atomicAdd(p + 3, v.w);
}

// ---------------- readout: out[g] = (gsum/cnt) @ W_out + b_out ----------------
__global__ __launch_bounds__(256)
void out_kernel(const float* __restrict__ gsum, const float* __restrict__ gcnt,
                const float* __restrict__ Wo, const float* __restrict__ bo,
                float* __restrict__ out, int nG) {
    long gw = ((long)blockIdx.x * blockDim.x + threadIdx.x) >> 5;
    if (gw >= nG) return;
    int lane = threadIdx.x & 31;
    float4 p = *(const float4*)(gsum + (size_t)gw * HH + lane * 4);
    float4 w = *(const float4*)(Wo + lane * 4);
    float s = p.x * w.x + p.y * w.y + p.z * w.z + p.w * w.w;
#pragma unroll
    for (int o = 16; o >= 1; o >>= 1) s += __shfl_xor(s, o, 32);
    if (lane == 0) {
        float c = fmaxf(gcnt[gw], 1.0f);
        out[gw] = s / c + bo[0];
    }
}

extern "C" void kernel_launch(void* const* d_in, const int* in_sizes, int n_in,
                              void* d_out, int out_size, void* d_ws, size_t ws_size,
                              hipStream_t stream) {
    const float* x       = (const float*)d_in[0];
    const int*   eidx    = (const int*)d_in[1];    // [2, E] flat
    const int*   batch   = (const int*)d_in[2];
    const float* W_embed = (const float*)d_in[3];
    const float* b_embed = (const float*)d_in[4];
    const float* W_gnn   = (const float*)d_in[5];  // [L,H,H]
    const float* b_gnn   = (const float*)d_in[6];  // [L,H]
    const float* gamma   = (const float*)d_in[7];
    const float* beta    = (const float*)d_in[8];
    const float* W_out   = (const float*)d_in[9];  // [H]
    const float* b_out   = (const float*)d_in[10];
    float* out = (float*)d_out;

    const int* src = eidx;
    const int* dst = eidx + EE;

    float* ws   = (float*)d_ws;
    float* dis  = ws;                         // N
    float* nrm  = dis + NN;                   // E
    float* h    = nrm + EE;                   // N*H
    float* m    = h   + (size_t)NN * HH;      // N*H
    float* acc  = m   + (size_t)NN * HH;      // N*H
    float* gsum = acc + (size_t)NN * HH;      // G*H
    float* gcnt = gsum + (size_t)GG * HH;     // G

    const int T = 256;

    // degrees (self loop => init 1.0), then dis = rsqrt(deg), edge norms
    fill_kernel<<<ceil_div(NN, T), T, 0, stream>>>(dis, 1.0f, NN);
    deg_kernel<<<ceil_div(EE, T), T, 0, stream>>>(dst, dis, EE);
    rsqrt_kernel<<<ceil_div(NN, T), T, 0, stream>>>(dis, NN);
    norm_kernel<<<ceil_div(EE, T), T, 0, stream>>>(src, dst, dis, nrm, EE);

    // embedding
    embed_kernel<<<ceil_div((long)NN * HH, T), T, 0, stream>>>(x, W_embed, b_embed, h, (long)NN * HH);

    // GCN layers
    dim3 ggrid(ceil_div(NN, 128), HH / 16, 1);
    long scatter_threads = ((long)EE + NN) * 32;
    for (int l = 0; l < LL; ++l) {
        gemm_kernel<<<ggrid, T, 0, stream>>>(h, W_gnn + (size_t)l * HH * HH, m, NN);
        biasinit_kernel<<<ceil_div((long)NN * HH, T), T, 0, stream>>>(acc, b_gnn + l * HH, (long)NN * HH);
        scatter_kernel<<<ceil_div(scatter_threads, T), T, 0, stream>>>(m, src, dst, nrm, dis, acc, EE, NN);
        ln_kernel<<<ceil_div((long)NN * 32, T), T, 0, stream>>>(h, acc, gamma + l * HH, beta + l * HH, NN);
    }

    // pool + readout
    fill_kernel<<<ceil_div((long)GG * HH + GG, T), T, 0, stream>>>(gsum, 0.0f, (long)GG * HH + GG);
    pool_kernel<<<ceil_div((long)NN * 32, T), T, 0, stream>>>(h, batch, gsum, gcnt, NN);
    out_kernel<<<ceil_div((long)GG * 32, T), T, 0, stream>>>(gsum, gcnt, W_out, b_out, out, GG);
}